// GAT_38740605010063
// MI455X (gfx1250) — compile-verified
//
#include <hip/hip_runtime.h>

#define NFEAT 128   // H*C
#define NH 2
#define NC 64
#define NEG_SLOPE 0.2f
#define BN_EPS 1e-5f

typedef float v2f __attribute__((ext_vector_type(2)));
typedef float v8f __attribute__((ext_vector_type(8)));

// LDS byte-offset of a pointer into a __shared__ array (as(3) ptrs are 32-bit)
__device__ __forceinline__ unsigned lds_off(const void* p) {
  return (unsigned)(size_t)(__attribute__((address_space(3))) const void*)p;
}

// ---------------------------------------------------------------------------
// Kernel 1: h = x @ W   ([N,128] x [128,128]) via fp32 WMMA 16x16x4.
// Block = 256 threads (8 waves). W (64KB) is staged into LDS once per block
// with GLOBAL_LOAD_ASYNC_TO_LDS_B128 (ASYNCcnt). Each wave then computes a
// 32-row x 128-col strip: 2 row-groups x 8 col-tiles = 16 C tiles (128 VGPRs),
// so every B fragment (ds_load_2addr from LDS) is reused by two WMMAs.
//
// Fragment layouts (wave32):
//  A (16x4 f32): v0: lanes0-15 K=0, lanes16-31 K=2; v1: K=1 / K=3.
//  B (4x16 f32): v0: lanes0-15 row K=0 (N=lane), lanes16-31 row K=2; v1: K=1/K=3.
//  C/D (16x16):  VGPR r: lanes0-15 -> M=r, lanes16-31 -> M=8+r, N=lane&15.
// ---------------------------------------------------------------------------
__global__ __launch_bounds__(256) void k_gemm_wmma(const float* __restrict__ x,
                                                   const float* __restrict__ W,
                                                   float* __restrict__ h,
                                                   int nStrips) {   // strips of 32 rows
  __shared__ float wlds[NFEAT * NFEAT];   // 64KB, row-major copy of W

  const int tid  = threadIdx.x;
  const int wave = tid >> 5;
  const int lane = tid & 31;

  // ---- async stage W into LDS: 4096 float4 transfers, 16 per thread ----
  {
    const unsigned base = lds_off(&wlds[0]);
#pragma unroll
    for (int i = 0; i < 16; ++i) {
      const int v4 = i * 256 + tid;                   // float4 index 0..4095
      const unsigned dsa = base + (unsigned)v4 * 16u; // LDS byte address
      const float* gp = W + (size_t)v4 * 4;
      asm volatile("global_load_async_to_lds_b128 %0, %1, off"
                   :: "v"(dsa), "v"(gp) : "memory");
    }
    asm volatile("s_wait_asynccnt 0x0" ::: "memory");
  }
  __syncthreads();

  const int strip = blockIdx.x * 8 + wave;
  if (strip < nStrips) {                   // wave-uniform: EXEC stays all-ones
    const int row0 = strip << 5;           // 32 rows per wave
    const int m    = lane & 15;
    const int ko   = (lane >> 4) << 1;     // 0 (lanes 0-15) or 2 (lanes 16-31)

    v8f c[2][8];
    const v8f vzero = {0.f, 0.f, 0.f, 0.f, 0.f, 0.f, 0.f, 0.f};
#pragma unroll
    for (int g = 0; g < 2; ++g)
#pragma unroll
      for (int i = 0; i < 8; ++i) c[g][i] = vzero;

    const float* xrow0 = x + (size_t)(row0 + m) * NFEAT;
    const float* xrow1 = xrow0 + 16 * NFEAT;

#pragma unroll 2
    for (int kt = 0; kt < 32; ++kt) {
      const int k0 = kt * 4;
      v2f a0, a1;
      a0.x = xrow0[k0 + ko];
      a0.y = xrow0[k0 + ko + 1];
      a1.x = xrow1[k0 + ko];
      a1.y = xrow1[k0 + ko + 1];
      const float* wb = &wlds[(k0 + ko) * NFEAT + m];   // LDS, row k0+ko
#pragma unroll
      for (int nt = 0; nt < 8; ++nt) {
        v2f b;
        b.x = wb[nt * 16];              // row k0+ko   -> offset0 = nt*16 dw
        b.y = wb[nt * 16 + NFEAT];      // row k0+ko+1 -> offset1 = nt*16+128 dw
#if __has_builtin(__builtin_amdgcn_wmma_f32_16x16x4_f32)
        c[0][nt] = __builtin_amdgcn_wmma_f32_16x16x4_f32(
            false, a0, false, b, (short)0, c[0][nt], false, false);
        c[1][nt] = __builtin_amdgcn_wmma_f32_16x16x4_f32(
            false, a1, false, b, (short)0, c[1][nt], false, false);
#else
        // parse-only fallback (host pass); never the perf path
#pragma unroll
        for (int r = 0; r < 8; ++r) {
          c[0][nt][r] += a0.x * b.x + a0.y * b.y;
          c[1][nt][r] += a1.x * b.x + a1.y * b.y;
        }
#endif
      }
    }

#pragma unroll
    for (int g = 0; g < 2; ++g) {
      const int rbase = row0 + g * 16;
#pragma unroll
      for (int nt = 0; nt < 8; ++nt)
#pragma unroll
        for (int r = 0; r < 8; ++r) {
          const int mm = (lane < 16) ? r : (r + 8);
          h[(size_t)(rbase + mm) * NFEAT + nt * 16 + m] = c[g][nt][r];
        }
    }
  }
}

// ---------------------------------------------------------------------------
// Kernel 2: per-(node,head) attention scalars a_src, a_dst
// ---------------------------------------------------------------------------
__global__ void k_att(const float* __restrict__ h, const float* __restrict__ att_src,
                      const float* __restrict__ att_dst, float* __restrict__ asrc,
                      float* __restrict__ adst, int n) {
  const int idx = blockIdx.x * blockDim.x + threadIdx.x;   // (node, head)
  if (idx >= n * NH) return;
  const int node = idx >> 1, hd = idx & 1;
  const float* hr = h + (size_t)node * NFEAT + hd * NC;
  const float* as = att_src + hd * NC;
  const float* ad = att_dst + hd * NC;
  float s = 0.f, d = 0.f;
#pragma unroll 8
  for (int c = 0; c < NC; ++c) { const float v = hr[c]; s += v * as[c]; d += v * ad[c]; }
  asrc[idx] = s;
  adst[idx] = d;
}

// ---------------------------------------------------------------------------
// Kernel 3: zero/init accumulators (d_out, segment max/sum, BN stats)
// ---------------------------------------------------------------------------
__global__ void k_init(float* __restrict__ out, float* __restrict__ maxv,
                       float* __restrict__ sums, float* __restrict__ colsum,
                       float* __restrict__ colsumsq, int n) {
  const int idx = blockIdx.x * blockDim.x + threadIdx.x;
  if (idx < n * NFEAT) out[idx] = 0.f;
  if (idx < n * NH) { maxv[idx] = -__builtin_inff(); sums[idx] = 0.f; }
  if (idx < NFEAT) { colsum[idx] = 0.f; colsumsq[idx] = 0.f; }
}

__device__ __forceinline__ void atomicMaxF(float* addr, float v) {
  if (v >= 0.f) atomicMax((int*)addr, __float_as_int(v));
  else          atomicMin((unsigned int*)addr, __float_as_uint(v));
}

// ---------------------------------------------------------------------------
// Kernel 4: edge logits (leaky relu) + segment max over dst (atomic)
// edges E..E+n-1 are the implicit self loops
// ---------------------------------------------------------------------------
__global__ void k_edge_max(const int* __restrict__ ei, const float* __restrict__ asrc,
                           const float* __restrict__ adst, float* __restrict__ el,
                           float* __restrict__ maxv, int E, int n) {
  const int e = blockIdx.x * blockDim.x + threadIdx.x;
  const int Et = E + n;
  if (e >= Et) return;
  const int s = (e < E) ? ei[e]     : (e - E);
  const int d = (e < E) ? ei[E + e] : (e - E);
#pragma unroll
  for (int hd = 0; hd < NH; ++hd) {
    float v = asrc[s * NH + hd] + adst[d * NH + hd];
    v = (v > 0.f) ? v : NEG_SLOPE * v;
    el[(size_t)e * NH + hd] = v;
    atomicMaxF(&maxv[d * NH + hd], v);
  }
}

// ---------------------------------------------------------------------------
// Kernel 5: p = exp(e - max[dst]); segment sum over dst (atomic); p in-place
// ---------------------------------------------------------------------------
__global__ void k_edge_exp(const int* __restrict__ ei, float* __restrict__ el,
                           const float* __restrict__ maxv, float* __restrict__ sums,
                           int E, int n) {
  const int e = blockIdx.x * blockDim.x + threadIdx.x;
  const int Et = E + n;
  if (e >= Et) return;
  const int d = (e < E) ? ei[E + e] : (e - E);
#pragma unroll
  for (int hd = 0; hd < NH; ++hd) {
    const float p = expf(el[(size_t)e * NH + hd] - maxv[d * NH + hd]);
    el[(size_t)e * NH + hd] = p;
    atomicAdd(&sums[d * NH + hd], p);
  }
}

// ---------------------------------------------------------------------------
// Kernel 6: weighted scatter  out[dst] += alpha * h[src].
// One wave per edge: lane -> 4 channels, single b128 gather of h[src] row.
// h and out both fit in the 192MB L2 -> random access is L2-resident.
// ---------------------------------------------------------------------------
__global__ __launch_bounds__(256) void k_aggregate(const int* __restrict__ ei,
                                                   const float* __restrict__ h,
                                                   const float* __restrict__ el,
                                                   const float* __restrict__ sums,
                                                   float* __restrict__ out,
                                                   int E, int n) {
  const int e = blockIdx.x * 8 + (threadIdx.x >> 5);
  const int lane = threadIdx.x & 31;
  const int Et = E + n;
  if (e >= Et) return;
  const int s = (e < E) ? ei[e]     : (e - E);
  const int d = (e < E) ? ei[E + e] : (e - E);
  const int c4 = lane << 2;            // channel base 0..124
  const int hd = lane >> 4;            // lanes 0-15 head0, 16-31 head1
  const float alpha = el[(size_t)e * NH + hd] / sums[d * NH + hd];
  const float4 hv = *reinterpret_cast<const float4*>(h + (size_t)s * NFEAT + c4);
  float* op = out + (size_t)d * NFEAT + c4;
  atomicAdd(op + 0, alpha * hv.x);
  atomicAdd(op + 1, alpha * hv.y);
  atomicAdd(op + 2, alpha * hv.z);
  atomicAdd(op + 3, alpha * hv.w);
}

// ---------------------------------------------------------------------------
// Kernel 7: bias + ReLU (in place) + per-column sum/sumsq for BN batch stats.
// Thread t owns column t; rows chunked 256 per block -> coalesced row loads.
// ---------------------------------------------------------------------------
__global__ __launch_bounds__(128) void k_relu_stats(float* __restrict__ out,
                                                    const float* __restrict__ bias,
                                                    float* __restrict__ colsum,
                                                    float* __restrict__ colsumsq,
                                                    int n) {
  const int col = threadIdx.x;
  const int r0 = blockIdx.x * 256;
  int rend = r0 + 256; if (rend > n) rend = n;
  const float b = bias[col];
  float s = 0.f, s2 = 0.f;
  for (int r = r0; r < rend; ++r) {
    float v = out[(size_t)r * NFEAT + col] + b;
    v = fmaxf(v, 0.f);
    out[(size_t)r * NFEAT + col] = v;
    s += v;
    s2 += v * v;
  }
  atomicAdd(&colsum[col], s);
  atomicAdd(&colsumsq[col], s2);
}

// ---------------------------------------------------------------------------
// Kernel 8: BatchNorm normalize (training-mode batch stats, biased var)
// ---------------------------------------------------------------------------
__global__ void k_bn(float* __restrict__ out, const float* __restrict__ colsum,
                     const float* __restrict__ colsumsq, const float* __restrict__ gamma,
                     const float* __restrict__ beta, int n) {
  const int idx = blockIdx.x * blockDim.x + threadIdx.x;
  if (idx >= n * NFEAT) return;
  const int col = idx & 127;
  const float invN = 1.f / (float)n;
  const float mean = colsum[col] * invN;
  const float var = colsumsq[col] * invN - mean * mean;
  const float v = out[idx];
  out[idx] = gamma[col] * (v - mean) * rsqrtf(var + BN_EPS) + beta[col];
}

extern "C" void kernel_launch(void* const* d_in, const int* in_sizes, int n_in,
                              void* d_out, int out_size, void* d_ws, size_t ws_size,
                              hipStream_t stream) {
  const float* x        = (const float*)d_in[0];
  const int*   ei       = (const int*)  d_in[1];   // [2,E]: row0=src, row1=dst
  const float* W        = (const float*)d_in[2];
  const float* att_src  = (const float*)d_in[3];
  const float* att_dst  = (const float*)d_in[4];
  const float* bias     = (const float*)d_in[5];
  const float* gamma    = (const float*)d_in[6];
  const float* beta     = (const float*)d_in[7];
  float* out = (float*)d_out;

  const int N = in_sizes[0] / NFEAT;
  const int E = in_sizes[1] / 2;
  const int Et = E + N;

  // workspace carve
  char* ws = (char*)d_ws;
  float* h        = (float*)ws; ws += (size_t)N * NFEAT * sizeof(float);
  float* asrc     = (float*)ws; ws += (size_t)N * NH * sizeof(float);
  float* adst     = (float*)ws; ws += (size_t)N * NH * sizeof(float);
  float* maxv     = (float*)ws; ws += (size_t)N * NH * sizeof(float);
  float* sums     = (float*)ws; ws += (size_t)N * NH * sizeof(float);
  float* el       = (float*)ws; ws += (size_t)Et * NH * sizeof(float);
  float* colsum   = (float*)ws; ws += NFEAT * sizeof(float);
  float* colsumsq = (float*)ws; ws += NFEAT * sizeof(float);

  // 1) GEMM (WMMA, W staged to LDS via async-to-LDS copies)
  const int nStrips = (N + 31) / 32;           // 32-row strips (N divisible)
  k_gemm_wmma<<<(nStrips + 7) / 8, 256, 0, stream>>>(x, W, h, nStrips);

  // 2) init accumulators
  {
    const int total = N * NFEAT;
    k_init<<<(total + 255) / 256, 256, 0, stream>>>(out, maxv, sums, colsum, colsumsq, N);
  }

  // 3) attention scalars
  k_att<<<(N * NH + 255) / 256, 256, 0, stream>>>(h, att_src, att_dst, asrc, adst, N);

  // 4) segment softmax over incoming edges per dst
  k_edge_max<<<(Et + 255) / 256, 256, 0, stream>>>(ei, asrc, adst, el, maxv, E, N);
  k_edge_exp<<<(Et + 255) / 256, 256, 0, stream>>>(ei, el, maxv, sums, E, N);

  // 5) weighted aggregation: one wave per edge
  k_aggregate<<<(Et + 7) / 8, 256, 0, stream>>>(ei, h, el, sums, out, E, N);

  // 6) bias + relu + batch stats, then BN normalize
  k_relu_stats<<<(N + 255) / 256, 128, 0, stream>>>(out, bias, colsum, colsumsq, N);
  k_bn<<<(N * NFEAT + 255) / 256, 256, 0, stream>>>(out, colsum, colsumsq, gamma, beta, N);
}